// PremadeRGCN_721554506567
// MI455X (gfx1250) — compile-verified
//
#include <hip/hip_runtime.h>
#include <hip/hip_bf16.h>

typedef __attribute__((ext_vector_type(2))) float v2f;
typedef __attribute__((ext_vector_type(8))) float v8f;

#define S_ 32
#define N_ 1024
#define E_ 8192
#define D_ 300
#define R_ 6
#define NG_ 10000
#define NS_ 25000
#define MAXE0 128

// ---------------- Kernel 1: W_r = sum_b comp[r,b] * basis[b] ----------------
__global__ void k_relweights(const float* __restrict__ comp,
                             const float* __restrict__ basis,
                             float* __restrict__ Wrel) {
    int idx = blockIdx.x * blockDim.x + threadIdx.x;
    if (idx >= R_ * D_ * D_) return;
    int r  = idx / (D_ * D_);
    int ij = idx % (D_ * D_);
    float acc = 0.f;
#pragma unroll
    for (int b = 0; b < R_; ++b)
        acc += comp[r * R_ + b] * basis[(size_t)b * D_ * D_ + ij];
    Wrel[idx] = acc;
}

// ---------------- Kernel 2: per-sample h0 (only node 0 matters) -------------
__global__ void k_h0(const float* __restrict__ x,
                     const int*   __restrict__ edge_index,
                     const int*   __restrict__ edge_type,
                     const float* __restrict__ Wrel,
                     const float* __restrict__ root,
                     const float* __restrict__ bias,
                     float* __restrict__ H0) {
    __shared__ int   s_src[MAXE0];
    __shared__ int   s_rel[MAXE0];
    __shared__ int   s_cnt;
    __shared__ float s_vec[D_];

    const int s = blockIdx.x;
    const int t = threadIdx.x;
    if (t == 0) s_cnt = 0;
    __syncthreads();

    const int* srcp = edge_index + (size_t)s * 2 * E_;
    const int* dstp = srcp + E_;
    const int* etp  = edge_type + (size_t)s * E_;

    for (int e = t; e < E_; e += blockDim.x) {
        if (dstp[e] == 0) {
            int k = atomicAdd(&s_cnt, 1);
            if (k < MAXE0) { s_src[k] = srcp[e]; s_rel[k] = etp[e]; }
        }
    }
    __syncthreads();
    const int cnt  = s_cnt;
    const int ncap = cnt < MAXE0 ? cnt : MAXE0;

    const float* xs = x + (size_t)s * N_ * D_;
    float agg = 0.f;
    for (int j = 0; j < ncap; ++j) {
        int srcn = s_src[j];
        int r    = s_rel[j];
        __syncthreads();
        for (int k = t; k < D_; k += blockDim.x)
            s_vec[k] = xs[(size_t)srcn * D_ + k];
        __syncthreads();
        if (t < D_) {
            const float* Wr = Wrel + (size_t)r * D_ * D_;
            float acc = 0.f;
            for (int k = 0; k < D_; ++k)
                acc = fmaf(s_vec[k], Wr[(size_t)k * D_ + t], acc);
            agg += acc;
        }
    }

    __syncthreads();
    for (int k = t; k < D_; k += blockDim.x) s_vec[k] = xs[k];  // x[s][0][:]
    __syncthreads();
    if (t < D_) {
        float rt = 0.f;
        for (int k = 0; k < D_; ++k)
            rt = fmaf(s_vec[k], root[(size_t)k * D_ + t], rt);
        float deg = cnt > 0 ? (float)cnt : 1.0f;
        float o = agg / deg + rt + bias[t];
        H0[s * D_ + t] = o > 0.f ? o : 0.f;
    }
}

// ---------------- Kernel 3: heads GEMM via V_WMMA_F32_16X16X4_F32 -----------
// logits[s, c] = sum_k W[c,k] * H0[s,k] + b[c]
// Block: 128 threads = 4 waves, each wave owns one 16-class tile, all 32 samples.
__global__ void k_heads(const float* __restrict__ W,
                        const float* __restrict__ bias,
                        const float* __restrict__ H0,
                        float* __restrict__ out,
                        int C) {
    __shared__ float sH0[S_ * D_];  // 38.4 KB
    for (int i = threadIdx.x; i < S_ * D_; i += blockDim.x)
        sH0[i] = H0[i];
    __syncthreads();

    const int lane = threadIdx.x & 31;
    const int wave = threadIdx.x >> 5;
    const int tile = blockIdx.x * 4 + wave;
    const int base = tile * 16;
    if (base >= C) return;  // uniform per wave

    const int mrow  = lane & 15;
    const int khalf = (lane >= 16) ? 2 : 0;
    int row = base + mrow;
    if (row > C - 1) row = C - 1;  // clamp loads; stores guarded below

    const float* Arow = W + (size_t)row * D_;
    const float* B0   = &sH0[(size_t)mrow * D_];
    const float* B1   = &sH0[(size_t)(16 + mrow) * D_];

    v8f acc0 = {};
    v8f acc1 = {};
    for (int k = 0; k < D_; k += 4) {
        __builtin_prefetch(&Arow[k + 32], 0, 0);  // global_prefetch ahead
        v2f a  = *(const v2f*)(&Arow[k + khalf]);
        v2f b0 = *(const v2f*)(&B0[k + khalf]);
        v2f b1 = *(const v2f*)(&B1[k + khalf]);
        acc0 = __builtin_amdgcn_wmma_f32_16x16x4_f32(
            false, a, false, b0, (short)0, acc0, false, false);
        acc1 = __builtin_amdgcn_wmma_f32_16x16x4_f32(
            false, a, false, b1, (short)0, acc1, false, false);
    }

    // C/D layout: vgpr i, lanes 0-15 -> (M=i, N=lane); lanes 16-31 -> (M=i+8, N=lane-16)
    const int moff  = (lane >= 16) ? 8 : 0;
    const int samp0 = lane & 15;        // acc0 -> samples 0..15
    const int samp1 = 16 + (lane & 15); // acc1 -> samples 16..31
#pragma unroll
    for (int i = 0; i < 8; ++i) {
        int cls = base + moff + i;
        if (cls < C) {
            float bv = bias[cls];
            out[(size_t)samp0 * C + cls] = acc0[i] + bv;
            out[(size_t)samp1 * C + cls] = acc1[i] + bv;
        }
    }
}

// ---------------- Kernel 4: per-(head,sample) max + logsumexp ---------------
__global__ void k_stats(const float* __restrict__ out_g,
                        const float* __restrict__ out_s,
                        float* __restrict__ maxs,
                        float* __restrict__ lses) {
    __shared__ float red[256];
    const int bx  = blockIdx.x;      // 0..63
    const int h   = bx >> 5;
    const int smp = bx & 31;
    const int t   = threadIdx.x;
    const int C   = h ? NS_ : NG_;
    const float* p = h ? (out_s + (size_t)smp * NS_) : (out_g + (size_t)smp * NG_);

    float m = -3.4e38f;
    for (int i = t; i < C; i += blockDim.x) m = fmaxf(m, p[i]);
    red[t] = m;
    __syncthreads();
    for (int s = 128; s > 0; s >>= 1) {
        if (t < s) red[t] = fmaxf(red[t], red[t + s]);
        __syncthreads();
    }
    m = red[0];
    __syncthreads();

    float sum = 0.f;
    for (int i = t; i < C; i += blockDim.x) sum += __expf(p[i] - m);
    red[t] = sum;
    __syncthreads();
    for (int s = 128; s > 0; s >>= 1) {
        if (t < s) red[t] += red[t + s];
        __syncthreads();
    }
    if (t == 0) { maxs[bx] = m; lses[bx] = __logf(red[0]); }
}

// ---------------- Kernel 5: finalize log_softmax in place -------------------
__global__ void k_final(float* __restrict__ out,
                        const float* __restrict__ maxs,
                        const float* __restrict__ lses) {
    const int total = S_ * NG_ + S_ * NS_;
    int idx = blockIdx.x * blockDim.x + threadIdx.x;
    if (idx >= total) return;
    int h, smp;
    if (idx < S_ * NG_) { h = 0; smp = idx / NG_; }
    else                { h = 1; smp = (idx - S_ * NG_) / NS_; }
    int si = h * 32 + smp;
    out[idx] = out[idx] - maxs[si] - lses[si];
}

extern "C" void kernel_launch(void* const* d_in, const int* in_sizes, int n_in,
                              void* d_out, int out_size, void* d_ws, size_t ws_size,
                              hipStream_t stream) {
    const float* x          = (const float*)d_in[0];
    const int*   edge_index = (const int*)d_in[1];
    const int*   edge_type  = (const int*)d_in[2];
    const float* basis      = (const float*)d_in[3];
    const float* comp       = (const float*)d_in[4];
    const float* root       = (const float*)d_in[5];
    const float* bias_conv  = (const float*)d_in[6];
    const float* Wg         = (const float*)d_in[7];
    const float* bg         = (const float*)d_in[8];
    const float* Ws         = (const float*)d_in[9];
    const float* bs         = (const float*)d_in[10];

    float* out   = (float*)d_out;
    float* out_g = out;                       // [S, NG]
    float* out_s = out + (size_t)S_ * NG_;    // [S, NS]

    float* ws   = (float*)d_ws;
    float* Wrel = ws;                         // R*D*D = 540000
    float* H0   = ws + 540000;                // S*D   = 9600
    float* maxs = ws + 549600;                // 64
    float* lses = ws + 549664;                // 64

    // 1) relation weights
    {
        int n = R_ * D_ * D_;
        k_relweights<<<(n + 255) / 256, 256, 0, stream>>>(comp, basis, Wrel);
    }
    // 2) h0 per sample
    k_h0<<<S_, 320, 0, stream>>>(x, edge_index, edge_type, Wrel, root, bias_conv, H0);
    // 3) heads GEMM (WMMA fp32)
    {
        int tiles_g = (NG_ + 15) / 16;   // 625
        int tiles_s = (NS_ + 15) / 16;   // 1563
        k_heads<<<(tiles_g + 3) / 4, 128, 0, stream>>>(Wg, bg, H0, out_g, NG_);
        k_heads<<<(tiles_s + 3) / 4, 128, 0, stream>>>(Ws, bs, H0, out_s, NS_);
    }
    // 4) softmax stats
    k_stats<<<64, 256, 0, stream>>>(out_g, out_s, maxs, lses);
    // 5) finalize
    {
        int total = S_ * NG_ + S_ * NS_;
        k_final<<<(total + 255) / 256, 256, 0, stream>>>(out, maxs, lses);
    }
}